// Watres_66443144069287
// MI455X (gfx1250) — compile-verified
//
#include <hip/hip_runtime.h>
#include <hip/hip_bf16.h>

#define TMAXN 43800
#define BATCH 256
#define NSPL  19
#define KPQ   6
#define SEQL  150

typedef __attribute__((ext_vector_type(2))) float v2f;
typedef __attribute__((ext_vector_type(8))) float v8f;

// ------------------------------------------------------------------
// Kernel 1: the two MLPs (batch=256, one thread per batch row, 1 block)
// BatchNorm needs cross-batch stats -> stage pre-BN activations in LDS.
// LDS staging buffer is capped at 256x64 floats (64 KB): layer 2's 128
// channels are processed in two 64-channel halves.
// ------------------------------------------------------------------
template<int DOUT>
__device__ void mlp_eval(const float* __restrict__ x,
                         const float* __restrict__ W1, const float* __restrict__ b1,
                         const float* __restrict__ g1, const float* __restrict__ be1,
                         const float* __restrict__ W2, const float* __restrict__ b2,
                         const float* __restrict__ g2, const float* __restrict__ be2,
                         const float* __restrict__ W3, const float* __restrict__ b3,
                         const float* __restrict__ W4, const float* __restrict__ b4,
                         float* __restrict__ outc,
                         float* Hbuf /*256*64*/, float* stat /*256*/)
{
    const int b = threadIdx.x;   // 0..255 = batch row

    float xr[32];
#pragma unroll
    for (int i = 0; i < 32; i++) xr[i] = x[b * 32 + i];

    // ----- layer 1: 32 -> 64, BN, relu -----
    float h1[64];
    for (int c = 0; c < 64; c++) {
        float s = b1[c];
#pragma unroll
        for (int i = 0; i < 32; i++) s += xr[i] * W1[i * 64 + c];
        h1[c] = s;
        Hbuf[b * 64 + c] = s;
    }
    __syncthreads();
    if (b < 64) {
        float s1 = 0.f, s2 = 0.f;
        for (int r = 0; r < 256; r++) { float v = Hbuf[r * 64 + b]; s1 += v; s2 += v * v; }
        float m = s1 * (1.f / 256.f);
        stat[b]       = m;
        stat[128 + b] = rsqrtf(s2 * (1.f / 256.f) - m * m + 1e-5f);
    }
    __syncthreads();
    for (int c = 0; c < 64; c++) {
        float hv = (h1[c] - stat[c]) * stat[128 + c] * g1[c] + be1[c];
        h1[c] = hv > 0.f ? hv : 0.f;
    }
    __syncthreads();

    // ----- layer 2: 64 -> 128, BN, relu (two 64-channel halves) -----
    float h2[128];
    for (int c = 0; c < 128; c++) {
        float s = b2[c];
#pragma unroll
        for (int i = 0; i < 64; i++) s += h1[i] * W2[i * 128 + c];
        h2[c] = s;
    }
    // first half: channels 0..63
    for (int c = 0; c < 64; c++) Hbuf[b * 64 + c] = h2[c];
    __syncthreads();
    if (b < 64) {
        float s1 = 0.f, s2 = 0.f;
        for (int r = 0; r < 256; r++) { float v = Hbuf[r * 64 + b]; s1 += v; s2 += v * v; }
        float m = s1 * (1.f / 256.f);
        stat[b]       = m;
        stat[128 + b] = rsqrtf(s2 * (1.f / 256.f) - m * m + 1e-5f);
    }
    __syncthreads();
    for (int c = 0; c < 64; c++) {
        float hv = (h2[c] - stat[c]) * stat[128 + c] * g2[c] + be2[c];
        h2[c] = hv > 0.f ? hv : 0.f;
    }
    // second half: channels 64..127 (sync separates stat reads above from rewrites)
    for (int c = 64; c < 128; c++) Hbuf[b * 64 + (c - 64)] = h2[c];
    __syncthreads();
    if (b < 64) {
        float s1 = 0.f, s2 = 0.f;
        for (int r = 0; r < 256; r++) { float v = Hbuf[r * 64 + b]; s1 += v; s2 += v * v; }
        float m = s1 * (1.f / 256.f);
        stat[b]       = m;
        stat[128 + b] = rsqrtf(s2 * (1.f / 256.f) - m * m + 1e-5f);
    }
    __syncthreads();
    for (int c = 64; c < 128; c++) {
        float hv = (h2[c] - stat[c - 64]) * stat[128 + (c - 64)] * g2[c] + be2[c];
        h2[c] = hv > 0.f ? hv : 0.f;
    }
    __syncthreads();

    // ----- layer 3 (leaky relu) fused with layer 4 accumulation -----
    float acc[DOUT];
#pragma unroll
    for (int o = 0; o < DOUT; o++) acc[o] = b4[o];
    for (int j = 0; j < SEQL; j++) {
        float s = b3[j];
#pragma unroll
        for (int i = 0; i < 128; i++) s += h2[i] * W3[i * SEQL + j];
        s = s > 0.f ? s : 0.01f * s;     // jax leaky_relu default slope 0.01
#pragma unroll
        for (int o = 0; o < DOUT; o++) acc[o] += s * W4[j * DOUT + o];
    }

    // ----- softmax over DOUT -----
    float mx = acc[0];
#pragma unroll
    for (int o = 1; o < DOUT; o++) mx = fmaxf(mx, acc[o]);
    float sum = 0.f;
#pragma unroll
    for (int o = 0; o < DOUT; o++) { acc[o] = __expf(acc[o] - mx); sum += acc[o]; }
    float inv = 1.f / sum;
#pragma unroll
    for (int o = 0; o < DOUT; o++) outc[b * DOUT + o] = acc[o] * inv;
}

__global__ __launch_bounds__(256) void k_mlp(
    const float* __restrict__ x,
    const float* wW1, const float* wb1, const float* wg1, const float* wbe1,
    const float* wW2, const float* wb2, const float* wg2, const float* wbe2,
    const float* wW3, const float* wb3, const float* wW4, const float* wb4,
    const float* pW1, const float* pb1, const float* pg1, const float* pbe1,
    const float* pW2, const float* pb2, const float* pg2, const float* pbe2,
    const float* pW3, const float* pb3, const float* pW4, const float* pb4,
    float* __restrict__ cwb, float* __restrict__ cpq)
{
    extern __shared__ float Hbuf[];       // 256*64 floats = 64 KB dynamic LDS
    __shared__ float stat[256];
    mlp_eval<NSPL>(x, wW1, wb1, wg1, wbe1, wW2, wb2, wg2, wbe2, wW3, wb3, wW4, wb4,
                   cwb, Hbuf, stat);
    __syncthreads();
    mlp_eval<KPQ>(x, pW1, pb1, pg1, pbe1, pW2, pb2, pg2, pbe2, pW3, pb3, pW4, pb4,
                  cpq, Hbuf, stat);
}

// ------------------------------------------------------------------
// Kernel 2: Jw = flip(J) * (coeffs_wb @ basis) via V_WMMA_F32_16X16X4_F32
// One wave per 16(M) x 16(N) tile; K = 19 padded to 20 = 5 WMMA steps.
// ------------------------------------------------------------------
__global__ __launch_bounds__(32) void k_wmma_jw(
    const float* __restrict__ cwb,    // (256, 19)
    const float* __restrict__ basis,  // (19, TMAX)
    const float* __restrict__ J,      // (256, TMAX)
    float* __restrict__ Jw)           // (256, TMAX)
{
    const int lane = threadIdx.x;            // 0..31
    const int t0   = blockIdx.x * 16;        // N tile (time)
    const int m0   = blockIdx.y * 16;        // M tile (batch)
    const int half = (lane < 16) ? 0 : 2;    // A/B K split between lane halves
    const int ml   = m0 + (lane & 15);
    int tl = t0 + (lane & 15);
    if (tl > TMAXN - 1) tl = TMAXN - 1;      // clamp loads; stores guarded below

    v8f c = {};
#pragma unroll
    for (int kk = 0; kk < 5; kk++) {
        const int k0 = kk * 4 + half;
        v2f a, bm;
        a.x  = (k0     < NSPL) ? cwb[ml * NSPL + k0    ] : 0.f;
        a.y  = (k0 + 1 < NSPL) ? cwb[ml * NSPL + k0 + 1] : 0.f;
        bm.x = (k0     < NSPL) ? basis[(k0    ) * TMAXN + tl] : 0.f;
        bm.y = (k0 + 1 < NSPL) ? basis[(k0 + 1) * TMAXN + tl] : 0.f;
        // D = A(16x4) * B(4x16) + C ; 8 args: neg_a, A, neg_b, B, c_mod, C, reuse_a, reuse_b
        c = __builtin_amdgcn_wmma_f32_16x16x4_f32(false, a, false, bm, (short)0, c,
                                                  false, false);
    }

    const int tcol = t0 + (lane & 15);
    if (tcol < TMAXN) {
        const int rbase = m0 + ((lane < 16) ? 0 : 8);
#pragma unroll
        for (int r = 0; r < 8; r++) {
            const int row = rbase + r;
            Jw[row * TMAXN + tcol] = J[row * TMAXN + (TMAXN - 1 - tcol)] * c[r];
        }
    }
}

// ------------------------------------------------------------------
// Kernel 3: per-row fused scan. Pass 1: cumsum(Jw) -> ST, capture ST[IDX],
// accumulate sum|pdf_a|. Pass 2: bit-identical rescan, build pQ, Chat,
// and the short cpQ prefix (IDX <= 6480).
// ------------------------------------------------------------------
__device__ __forceinline__ float block_iscan(float v, float* wsum, float* total)
{
    const int lane = threadIdx.x & 31;
    const int wid  = threadIdx.x >> 5;      // 8 wave32 per 256-thread block
#pragma unroll
    for (int d = 1; d < 32; d <<= 1) {
        float n = __shfl_up(v, d, 32);
        if (lane >= d) v += n;
    }
    if (lane == 31) wsum[wid] = v;
    __syncthreads();
    float off = 0.f, tot = 0.f;
#pragma unroll
    for (int w = 0; w < 8; w++) {
        float s = wsum[w];
        tot += s;
        if (w < wid) off += s;
    }
    __syncthreads();
    *total = tot;
    return v + off;
}

__device__ __forceinline__ float block_reduce(float v, float* red)
{
    const int tid = threadIdx.x;
    red[tid] = v;
    __syncthreads();
#pragma unroll
    for (int s = 128; s > 0; s >>= 1) {
        if (tid < s) red[tid] += red[tid + s];
        __syncthreads();
    }
    float r = red[0];
    __syncthreads();
    return r;
}

// gamma pdfs for a in {1,2,3,4,6}, b=1: coeff = 1/Gamma(a)
__device__ __forceinline__ void gamma_pdfs(float xg, float jw, float p[5])
{
    if (xg > 0.f) {
        float e  = __expf(-xg) * jw;
        float x2 = xg * xg;
        p[0] = e;
        p[1] = e * xg;
        p[2] = e * x2 * 0.5f;
        p[3] = e * x2 * xg * (1.f / 6.f);
        p[4] = e * x2 * x2 * xg * (1.f / 120.f);
    } else {
        p[0] = p[1] = p[2] = p[3] = p[4] = 0.f;
    }
}

__global__ __launch_bounds__(256) void k_scan(
    const float* __restrict__ Jw,   // (256, TMAX)
    const float* __restrict__ cpq,  // (256, 6)
    const float* __restrict__ CJ,   // (256, TMAX)
    float* __restrict__ out)        // [Chat 256][ST_idx 256*9][cpQ_idx 256*9]
{
    __shared__ float wsum[8];
    __shared__ float red[256];
    __shared__ float gco[5];

    const int b   = blockIdx.x;
    const int tid = threadIdx.x;
    const float* row = Jw + (size_t)b * TMAXN;
    const int NCH = (TMAXN + 255) / 256;       // 172 chunks

    // ---- pass 1: ST scan + |pdf| sums ----
    float carry = 0.f;
    float sacc[5] = {0.f, 0.f, 0.f, 0.f, 0.f};
    for (int ch = 0; ch < NCH; ch++) {
        const int t = ch * 256 + tid;
        float v = (t < TMAXN) ? row[t] : 0.f;
        if (t + 256 < TMAXN) __builtin_prefetch(&row[t + 256], 0, 1);
        float tot;
        float st = carry + block_iscan(v, wsum, &tot);
        carry += tot;
        if (t < TMAXN) {
            if (t >= 720 && t <= 6480 && (t % 720) == 0)
                out[BATCH + b * 9 + (t / 720 - 1)] = st;          // ST[:, IDX]
            float p[5];
            gamma_pdfs(st * (1.f / 2000.f), v, p);
#pragma unroll
            for (int k = 0; k < 5; k++) sacc[k] += fabsf(p[k]);
        }
    }
    for (int k = 0; k < 5; k++) {
        float s = block_reduce(sacc[k], red);
        if (tid == 0) gco[k] = cpq[b * KPQ + k] / fmaxf(s, 1e-12f);
    }
    __syncthreads();
    const float g0 = gco[0], g1 = gco[1], g2 = gco[2], g3 = gco[3], g4 = gco[4];

    // ---- pass 2: identical rescan -> pQ, Chat, cpQ prefix ----
    carry = 0.f;
    float qcarry = 0.f;
    float chat = 0.f;
    for (int ch = 0; ch < NCH; ch++) {
        const int t = ch * 256 + tid;
        float v = (t < TMAXN) ? row[t] : 0.f;
        float tot;
        float st = carry + block_iscan(v, wsum, &tot);
        carry += tot;
        float p[5];
        gamma_pdfs(st * (1.f / 2000.f), v, p);
        float pq = g0 * p[0] + g1 * p[1] + g2 * p[2] + g3 * p[3] + g4 * p[4];
        if (t >= TMAXN) pq = 0.f;
        if (t < TMAXN) chat += CJ[(size_t)b * TMAXN + (TMAXN - 1 - t)] * pq;
        if (ch * 256 <= 6480) {                 // uniform branch; cpQ only sampled <= 6480
            float qtot;
            float q = qcarry + block_iscan(pq, wsum, &qtot);
            qcarry += qtot;
            if (t >= 720 && t <= 6480 && (t % 720) == 0)
                out[BATCH + BATCH * 9 + b * 9 + (t / 720 - 1)] = q;   // cpQ[:, IDX]
        }
    }
    chat = block_reduce(chat, red);
    if (tid == 0) out[b] = chat;
}

// ------------------------------------------------------------------
extern "C" void kernel_launch(void* const* d_in, const int* in_sizes, int n_in,
                              void* d_out, int out_size, void* d_ws, size_t ws_size,
                              hipStream_t stream)
{
    const float* x     = (const float*)d_in[0];
    const float* J     = (const float*)d_in[1];
    const float* CJ    = (const float*)d_in[2];
    const float* basis = (const float*)d_in[3];

    float* ws  = (float*)d_ws;
    float* cwb = ws;                                 // 256*19
    float* cpq = cwb + BATCH * NSPL;                 // 256*6
    float* Jw  = cpq + BATCH * KPQ;                  // 256*43800 (~44.9 MB, L2-resident)
    float* outf = (float*)d_out;

    k_mlp<<<1, 256, 256 * 64 * sizeof(float), stream>>>(
        x,
        (const float*)d_in[4],  (const float*)d_in[5],  (const float*)d_in[6],  (const float*)d_in[7],
        (const float*)d_in[8],  (const float*)d_in[9],  (const float*)d_in[10], (const float*)d_in[11],
        (const float*)d_in[12], (const float*)d_in[13], (const float*)d_in[14], (const float*)d_in[15],
        (const float*)d_in[16], (const float*)d_in[17], (const float*)d_in[18], (const float*)d_in[19],
        (const float*)d_in[20], (const float*)d_in[21], (const float*)d_in[22], (const float*)d_in[23],
        (const float*)d_in[24], (const float*)d_in[25], (const float*)d_in[26], (const float*)d_in[27],
        cwb, cpq);

    dim3 g2((TMAXN + 15) / 16, BATCH / 16);          // 2738 x 16 tiles
    k_wmma_jw<<<g2, 32, 0, stream>>>(cwb, basis, J, Jw);

    k_scan<<<BATCH, 256, 0, stream>>>(Jw, cpq, CJ, outf);
}